// VectorQuantizer_91276644974784
// MI455X (gfx1250) — compile-verified
//
#include <hip/hip_runtime.h>

typedef __attribute__((ext_vector_type(2))) float v2f;
typedef __attribute__((ext_vector_type(4))) float v4f;
typedef __attribute__((ext_vector_type(8))) float v8f;

#define NUM_K   4096
#define DIM     128
#define NROW    32768                 // 32 * 32 * 32 flattened vectors
#define HW      1024                  // 32 * 32
#define Q_SIZE  4194304               // 32*128*32*32
#define LOSS_OFF 4194304
#define PERP_OFF 4194305
#define ENC_OFF  4194306ull
#define IDX_OFF  138412034ull         // ENC_OFF + 32768*4096

// -------- workspace layout (floats) --------
// [0, 4096)       : w_sq
// [4096, 8192)    : counts
// [8192]          : loss accumulator
// ints at float-offset 8448: 32768 encoding indices

__global__ void vq_init_kernel(float* __restrict__ wsf) {
    int i = blockIdx.x * blockDim.x + threadIdx.x;
    if (i < NUM_K + 1) wsf[NUM_K + i] = 0.0f;   // counts + loss accum
}

__global__ void vq_wsq_kernel(const float* __restrict__ W, float* __restrict__ wsq) {
    int k = blockIdx.x * blockDim.x + threadIdx.x;
    if (k >= NUM_K) return;
    const float4* row = (const float4*)(W + (size_t)k * DIM);
    float s = 0.0f;
    #pragma unroll 8
    for (int j = 0; j < DIM / 4; ++j) {
        float4 v = row[j];
        s += v.x * v.x + v.y * v.y + v.z * v.z + v.w * v.w;
    }
    wsq[k] = s;
}

// Block: 128 threads (4 waves). One block handles 32 consecutive flattened rows
// as TWO 16-row A-tiles so every B fragment feeds two WMMAs.
// Wave w owns code stripe [c*64 + w*16, +16) for chunk c in 0..63.
__global__ void __launch_bounds__(128)
vq_main_kernel(const float* __restrict__ x,      // [32,128,32,32] NCHW
               const float* __restrict__ W,      // [4096,128]
               const float* __restrict__ wsq,    // [4096]
               float* __restrict__ out,          // full output buffer
               float* __restrict__ counts,
               float* __restrict__ lossacc,
               int* __restrict__ idxbuf) {
    __shared__ float lds_x[32 * 132];   // 32 rows, padded stride 132
    __shared__ float sbv[4][32];
    __shared__ int   sbi[4][32];
    __shared__ int   fidx[32];

    const int tid  = threadIdx.x;
    const int n0   = blockIdx.x * 32;           // first row of the tile
    const int b    = n0 / HW;                   // batch image
    const int hw0  = n0 % HW;                   // 32-aligned -> contiguous
    const float* xin = x + (size_t)b * DIM * HW;

    // ---- stage x tile [32 rows][128 dims] into LDS (transpose gather) ----
    {
        const float* base = xin + (size_t)tid * HW + hw0;   // dim = tid
        #pragma unroll
        for (int g = 0; g < 8; ++g) {
            float4 r = *(const float4*)(base + g * 4);
            lds_x[(g * 4 + 0) * 132 + tid] = r.x;
            lds_x[(g * 4 + 1) * 132 + tid] = r.y;
            lds_x[(g * 4 + 2) * 132 + tid] = r.z;
            lds_x[(g * 4 + 3) * 132 + tid] = r.w;
        }
    }
    __syncthreads();

    const int wave = tid >> 5;
    const int lane = tid & 31;
    const int l16  = lane & 15;
    const int half = lane >> 4;          // 0: rows 0-7 / K 0,1 ; 1: rows 8-15 / K 2,3
    const int o0   = half ? 2 : 0;

    // ||x||^2 is constant per row: it cannot change the argmin and distances
    // are never emitted, so score := w_sq[k] - 2 * <x,w_k> is sufficient.
    float bv0[8], bv1[8];
    int   bi0[8], bi1[8];
    #pragma unroll
    for (int r = 0; r < 8; ++r) {
        bv0[r] = 3.4e38f; bi0[r] = 0;
        bv1[r] = 3.4e38f; bi1[r] = 0;
    }

    // ---- main loop: 64 chunks of 64 codes (16 per wave) ----
    for (int c = 0; c < 64; ++c) {
        const int code = c * 64 + wave * 16 + l16;
        const float* wrow = W + (size_t)code * DIM;

        v8f acc0 = {};
        v8f acc1 = {};
        #pragma unroll 8
        for (int kd = 0; kd < DIM; kd += 4) {
            float2 bvv = *(const float2*)(wrow + kd + o0);
            float2 a0v = *(const float2*)&lds_x[l16 * 132 + kd + o0];
            float2 a1v = *(const float2*)&lds_x[(16 + l16) * 132 + kd + o0];
            v2f bb; bb.x = bvv.x; bb.y = bvv.y;
            v2f a0; a0.x = a0v.x; a0.y = a0v.y;
            v2f a1; a1.x = a1v.x; a1.y = a1v.y;
            acc0 = __builtin_amdgcn_wmma_f32_16x16x4_f32(
                false, a0, false, bb, (short)0, acc0, false, false);
            acc1 = __builtin_amdgcn_wmma_f32_16x16x4_f32(
                false, a1, false, bb, (short)0, acc1, false, false);
        }

        const float wk = wsq[code];
        #pragma unroll
        for (int r = 0; r < 8; ++r) {
            float s0 = __builtin_fmaf(-2.0f, acc0[r], wk);
            float s1 = __builtin_fmaf(-2.0f, acc1[r], wk);
            bool t0 = s0 < bv0[r];                // strict: keeps first index
            bool t1 = s1 < bv1[r];
            bv0[r] = t0 ? s0 : bv0[r];
            bi0[r] = t0 ? code : bi0[r];
            bv1[r] = t1 ? s1 : bv1[r];
            bi1[r] = t1 ? code : bi1[r];
        }
    }

    // ---- once-only argmin across the 16 lanes of each half-wave ----
    #pragma unroll
    for (int r = 0; r < 8; ++r) {
        float v0 = bv0[r]; int i0 = bi0[r];
        float v1 = bv1[r]; int i1 = bi1[r];
        #pragma unroll
        for (int m = 1; m <= 8; m <<= 1) {
            float ov0 = __shfl_xor(v0, m, 32); int oi0 = __shfl_xor(i0, m, 32);
            float ov1 = __shfl_xor(v1, m, 32); int oi1 = __shfl_xor(i1, m, 32);
            bool t0 = (ov0 < v0) | ((ov0 == v0) & (oi0 < i0));
            bool t1 = (ov1 < v1) | ((ov1 == v1) & (oi1 < i1));
            v0 = t0 ? ov0 : v0; i0 = t0 ? oi0 : i0;
            v1 = t1 ? ov1 : v1; i1 = t1 ? oi1 : i1;
        }
        bv0[r] = v0; bi0[r] = i0;
        bv1[r] = v1; bi1[r] = i1;
    }

    // ---- cross-wave argmin combine ----
    if (l16 == 0) {
        #pragma unroll
        for (int r = 0; r < 8; ++r) {
            sbv[wave][half * 8 + r] = bv0[r];
            sbi[wave][half * 8 + r] = bi0[r];
            sbv[wave][16 + half * 8 + r] = bv1[r];
            sbi[wave][16 + half * 8 + r] = bi1[r];
        }
    }
    __syncthreads();
    if (tid < 32) {
        float bvx = sbv[0][tid];
        int   bix = sbi[0][tid];
        #pragma unroll
        for (int w = 1; w < 4; ++w) {
            float v = sbv[w][tid];
            int   i = sbi[w][tid];
            bool t = (v < bvx) | ((v == bvx) & (i < bix));
            bvx = t ? v : bvx;
            bix = t ? i : bix;
        }
        fidx[tid] = bix;
        idxbuf[n0 + tid] = bix;
        out[IDX_OFF + (size_t)(n0 + tid)] = (float)bix;
        atomicAdd(&counts[bix], 1.0f);
    }
    __syncthreads();

    // ---- quantized output (NCHW) + MSE partial. thread tid owns dim d = tid ----
    {
        const int d = tid;
        float lsum = 0.0f;
        float* qbase = out + (size_t)b * DIM * HW + (size_t)d * HW + hw0;
        #pragma unroll
        for (int rg = 0; rg < 8; ++rg) {
            const int i0 = fidx[rg * 4 + 0];
            const int i1 = fidx[rg * 4 + 1];
            const int i2 = fidx[rg * 4 + 2];
            const int i3 = fidx[rg * 4 + 3];
            v4f q;
            q.x = W[(size_t)i0 * DIM + d];
            q.y = W[(size_t)i1 * DIM + d];
            q.z = W[(size_t)i2 * DIM + d];
            q.w = W[(size_t)i3 * DIM + d];
            float d0 = q.x - lds_x[(rg * 4 + 0) * 132 + d];
            float d1 = q.y - lds_x[(rg * 4 + 1) * 132 + d];
            float d2 = q.z - lds_x[(rg * 4 + 2) * 132 + d];
            float d3 = q.w - lds_x[(rg * 4 + 3) * 132 + d];
            lsum += d0 * d0 + d1 * d1 + d2 * d2 + d3 * d3;
            // streamed output, never re-read on device -> non-temporal
            __builtin_nontemporal_store(q, (v4f*)(qbase + rg * 4));
        }
        #pragma unroll
        for (int m = 16; m >= 1; m >>= 1) lsum += __shfl_xor(lsum, m, 32);
        if (lane == 0) atomicAdd(lossacc, lsum);
    }
}

// One-hot encodings: 537 MB of float2 stores (ENC_OFF is 8B-aligned, not 16B).
// Pure write stream, cannot fit in L2 -> non-temporal stores keep the weight
// matrix and quantized tiles resident in L2 for the other kernels.
__global__ void vq_onehot_kernel(const int* __restrict__ idxbuf, float* __restrict__ out) {
    v2f* enc = (v2f*)(out + ENC_OFF);
    const size_t total2 = (size_t)NROW * (NUM_K / 2);
    const size_t stride = (size_t)gridDim.x * blockDim.x;
    for (size_t i = (size_t)blockIdx.x * blockDim.x + threadIdx.x; i < total2; i += stride) {
        const size_t n  = i >> 11;          // 2048 float2 per row
        const int    k2 = (int)(i & 2047);
        const int    id = idxbuf[n];
        v2f v; v.x = 0.0f; v.y = 0.0f;
        if ((id >> 1) == k2) { if (id & 1) v.y = 1.0f; else v.x = 1.0f; }
        __builtin_nontemporal_store(v, &enc[i]);
    }
}

__global__ void vq_finalize_kernel(const float* __restrict__ counts,
                                   const float* __restrict__ lossacc,
                                   float* __restrict__ out) {
    __shared__ float red[256];
    const int tid = threadIdx.x;
    float s = 0.0f;
    for (int k = tid; k < NUM_K; k += 256) {
        float p = counts[k] * (1.0f / (float)NROW);
        s += p * __logf(p + 1e-10f);
    }
    red[tid] = s;
    __syncthreads();
    for (int off = 128; off > 0; off >>= 1) {
        if (tid < off) red[tid] += red[tid + off];
        __syncthreads();
    }
    if (tid == 0) {
        out[PERP_OFF] = __expf(-red[0]);
        float loss = 1.25f * lossacc[0] * (1.0f / (float)Q_SIZE);
        if (isnan(loss) || isinf(loss)) loss = 0.1f;
        out[LOSS_OFF] = loss;
    }
}

extern "C" void kernel_launch(void* const* d_in, const int* in_sizes, int n_in,
                              void* d_out, int out_size, void* d_ws, size_t ws_size,
                              hipStream_t stream) {
    const float* x = (const float*)d_in[0];   // [32,128,32,32] fp32
    const float* W = (const float*)d_in[1];   // [4096,128] fp32
    float* out = (float*)d_out;

    float* wsf     = (float*)d_ws;
    float* wsq     = wsf;                 // 4096
    float* counts  = wsf + 4096;          // 4096
    float* lossacc = wsf + 8192;          // 1
    int*   idxbuf  = (int*)(wsf + 8448);  // 32768

    vq_init_kernel<<<(NUM_K + 1 + 255) / 256, 256, 0, stream>>>(wsf);
    vq_wsq_kernel<<<NUM_K / 256, 256, 0, stream>>>(W, wsq);
    vq_main_kernel<<<NROW / 32, 128, 0, stream>>>(x, W, wsq, out, counts, lossacc, idxbuf);
    vq_onehot_kernel<<<8192, 256, 0, stream>>>(idxbuf, out);
    vq_finalize_kernel<<<1, 256, 0, stream>>>(counts, lossacc, out);
}